// _Interactions_79791902425118
// MI455X (gfx1250) — compile-verified
//
#include <hip/hip_runtime.h>
#include <math.h>

#define DD 64

typedef __attribute__((ext_vector_type(2))) float v2f;
typedef __attribute__((ext_vector_type(8))) float v8f;

// ---------------------------------------------------------------------------
// GEMM: Out[nrows x 64] = A[nrows x 64] @ W[64 x 64]  (+ optional bias, relu)
// fp32 WMMA path: V_WMMA_F32_16X16X4_F32, one wave computes 16 rows x 64 cols.
// W is stored in LDS pair-interleaved: sW2[kp][col] = {W[2kp][col], W[2kp+1][col]}
// so each B fragment is a single aligned ds_load_b64 into an even VGPR pair.
// ---------------------------------------------------------------------------
__global__ __launch_bounds__(256) void gemm64_wmma(
    const float* __restrict__ A, const float* __restrict__ W,
    const float* __restrict__ bias, float* __restrict__ Out,
    int nrows, int fuse_bias_relu)
{
    __shared__ float2 sW2[(DD / 2) * DD];   // 16 KB
    const int tid = threadIdx.x;
    // cooperative fill: coalesced b128 reads, pair-interleaved b32 DS scatter
    for (int i = tid; i < DD * DD / 4; i += 256) {
        const int r  = i >> 4;              // source row 0..63
        const int c4 = (i & 15) * 4;        // col chunk
        float4 w4 = *(const float4*)&W[r * DD + c4];
        const int kp = r >> 1, comp = r & 1;
        float* base = (float*)&sW2[kp * DD + c4];
        base[0 * 2 + comp] = w4.x;
        base[1 * 2 + comp] = w4.y;
        base[2 * 2 + comp] = w4.z;
        base[3 * 2 + comp] = w4.w;
    }
    __syncthreads();

    const int wave = tid >> 5;
    const int lane = tid & 31;
    const int row0 = (blockIdx.x * 8 + wave) * 16;
    if (row0 >= nrows) return;              // wave-uniform; active waves keep EXEC all-1

    const int half = lane >> 4;             // 0: lanes 0-15, 1: lanes 16-31
    const int lr   = lane & 15;

    const float* __restrict__ arow = A + (size_t)(row0 + lr) * DD;
    const float2* __restrict__ bbase = &sW2[half * DD + lr];  // kp = k0/2 + half

    v8f c0 = {}, c1 = {}, c2 = {}, c3 = {};
    #pragma unroll
    for (int k0 = 0; k0 < DD; k0 += 4) {
        v2f a;
        a.x = arow[k0 + 2 * half];          // A 16x4 layout: VGPR0={K0,K2}, VGPR1={K1,K3}
        a.y = arow[k0 + 2 * half + 1];
        const float2* bp = bbase + (k0 >> 1) * DD;
        v2f b0 = *(const v2f*)&bp[0];
        v2f b1 = *(const v2f*)&bp[16];
        v2f b2 = *(const v2f*)&bp[32];
        v2f b3 = *(const v2f*)&bp[48];
        c0 = __builtin_amdgcn_wmma_f32_16x16x4_f32(false, a, false, b0, (short)0, c0, false, false);
        c1 = __builtin_amdgcn_wmma_f32_16x16x4_f32(false, a, false, b1, (short)0, c1, false, false);
        c2 = __builtin_amdgcn_wmma_f32_16x16x4_f32(false, a, false, b2, (short)0, c2, false, false);
        c3 = __builtin_amdgcn_wmma_f32_16x16x4_f32(false, a, false, b3, (short)0, c3, false, false);
    }

    float bb0 = 0.f, bb1 = 0.f, bb2 = 0.f, bb3 = 0.f;
    if (fuse_bias_relu) {
        bb0 = bias[lr]; bb1 = bias[lr + 16]; bb2 = bias[lr + 32]; bb3 = bias[lr + 48];
    }
    #pragma unroll
    for (int i = 0; i < 8; ++i) {
        const int m = i + 8 * half;         // C/D layout: VGPR i -> row i (+8 for hi half)
        float* __restrict__ o = Out + (size_t)(row0 + m) * DD + lr;
        float v0 = c0[i], v1 = c1[i], v2 = c2[i], v3 = c3[i];
        if (fuse_bias_relu) {
            v0 = fmaxf(v0 + bb0, 0.f);
            v1 = fmaxf(v1 + bb1, 0.f);
            v2 = fmaxf(v2 + bb2, 0.f);
            v3 = fmaxf(v3 + bb3, 0.f);
        }
        o[0] = v0; o[16] = v1; o[32] = v2; o[48] = v3;
    }
}

// ---------------------------------------------------------------------------
// Per-node attention scalars + per-layer state init.
// ---------------------------------------------------------------------------
__global__ __launch_bounds__(256) void rowprep(
    const float* __restrict__ hx, const float* __restrict__ a_src,
    const float* __restrict__ a_dst,
    float* __restrict__ ssrc, float* __restrict__ sdst,
    float* __restrict__ mx, float* __restrict__ ssum,
    float* __restrict__ agg, int n)
{
    int i = blockIdx.x * blockDim.x + threadIdx.x;
    if (i >= n) return;
    const float* r = hx + (size_t)i * DD;
    float s0 = 0.f, s1 = 0.f;
    #pragma unroll
    for (int k = 0; k < DD; k += 4) {
        float4 h4  = *(const float4*)&r[k];
        float4 as4 = *(const float4*)&a_src[k];
        float4 ad4 = *(const float4*)&a_dst[k];
        s0 += h4.x * as4.x + h4.y * as4.y + h4.z * as4.z + h4.w * as4.w;
        s1 += h4.x * ad4.x + h4.y * ad4.y + h4.z * ad4.z + h4.w * ad4.w;
    }
    ssrc[i] = s0;
    sdst[i] = s1;
    mx[i]   = -__builtin_inff();
    ssum[i] = 0.f;
    float4 z = {0.f, 0.f, 0.f, 0.f};
    float* ar = agg + (size_t)i * DD;
    #pragma unroll
    for (int k = 0; k < DD; k += 4) *(float4*)&ar[k] = z;
}

// float atomic max via sign-aware int/uint atomics (valid with -inf init)
__device__ __forceinline__ void atomicMaxFloat(float* addr, float val)
{
    if (val >= 0.f) atomicMax((int*)addr, __float_as_int(val));
    else            atomicMin((unsigned int*)addr, __float_as_uint(val));
}

__global__ __launch_bounds__(256) void edge_alpha(
    const int* __restrict__ src, const int* __restrict__ dst,
    const float* __restrict__ ssrc, const float* __restrict__ sdst,
    float* __restrict__ ebuf, float* __restrict__ mx, int E)
{
    int e = blockIdx.x * blockDim.x + threadIdx.x;
    if (e >= E) return;
    int s = src[e], d = dst[e];
    float a = ssrc[s] + sdst[d];
    a = (a > 0.f) ? a : 0.2f * a;       // leaky relu, slope 0.2
    ebuf[e] = a;
    atomicMaxFloat(&mx[d], a);
}

__global__ __launch_bounds__(256) void edge_exp(
    const int* __restrict__ dst, float* __restrict__ ebuf,
    const float* __restrict__ mx, float* __restrict__ ssum, int E)
{
    int e = blockIdx.x * blockDim.x + threadIdx.x;
    if (e >= E) return;
    int d = dst[e];
    float v = __expf(ebuf[e] - mx[d]);
    ebuf[e] = v;
    atomicAdd(&ssum[d], v);
}

// one wave per edge, 2 features per lane
__global__ __launch_bounds__(256) void edge_scatter(
    const int* __restrict__ src, const int* __restrict__ dst,
    const float* __restrict__ ebuf, const float* __restrict__ ssum,
    const float* __restrict__ hx, float* __restrict__ agg, int E)
{
    int e    = (int)((blockIdx.x * (unsigned)blockDim.x + threadIdx.x) >> 5);
    int lane = threadIdx.x & 31;
    if (e >= E) return;
    int s = src[e], d = dst[e];
    float p = ebuf[e] / ssum[d];
    float2 h2 = *(const float2*)&hx[(size_t)s * DD + lane * 2];
    atomicAdd(&agg[(size_t)d * DD + lane * 2 + 0], h2.x * p);
    atomicAdd(&agg[(size_t)d * DD + lane * 2 + 1], h2.y * p);
}

__global__ __launch_bounds__(256) void finalize(
    const float* __restrict__ hprev, const float* __restrict__ agg,
    const float* __restrict__ bias, float* __restrict__ out, int total)
{
    int i = blockIdx.x * blockDim.x + threadIdx.x;
    if (i >= total) return;
    float v = agg[i] + bias[i & (DD - 1)];
    out[i] = hprev[i] + fmaxf(v, 0.f);
}

// ---------------------------------------------------------------------------
extern "C" void kernel_launch(void* const* d_in, const int* in_sizes, int n_in,
                              void* d_out, int out_size, void* d_ws, size_t ws_size,
                              hipStream_t stream)
{
    const float* x   = (const float*)d_in[0];
    const int*   ei  = (const int*)  d_in[1];   // edge_index [2, E]
    const float* W0  = (const float*)d_in[4];
    const float* b0  = (const float*)d_in[5];
    const float* W1  = (const float*)d_in[6];
    const float* as1 = (const float*)d_in[7];
    const float* ad1 = (const float*)d_in[8];
    const float* b1  = (const float*)d_in[9];
    const float* W2  = (const float*)d_in[10];
    const float* as2 = (const float*)d_in[11];
    const float* ad2 = (const float*)d_in[12];
    const float* b2  = (const float*)d_in[13];

    const int N = in_sizes[0] / DD;
    const int E = in_sizes[2];          // edge_weight length
    const int* src = ei;
    const int* dst = ei + E;

    float* out = (float*)d_out;

    // workspace carve-up (~42 MB of floats)
    float* ws   = (float*)d_ws;
    float* h0   = ws;  ws += (size_t)N * DD;
    float* hx   = ws;  ws += (size_t)N * DD;
    float* agg  = ws;  ws += (size_t)N * DD;
    float* ssrc = ws;  ws += N;
    float* sdst = ws;  ws += N;
    float* mx   = ws;  ws += N;
    float* ssum = ws;  ws += N;
    float* ebuf = ws;  ws += E;

    const int gemmBlocks = ((N + 15) / 16 + 7) / 8;   // 8 waves x 16 rows per block
    dim3 b256(256);

    // layer 0: h0 = relu(x @ W0 + b0)
    gemm64_wmma<<<gemmBlocks, b256, 0, stream>>>(x, W0, b0, h0, N, 1);

    const float* hin = h0;
    for (int layer = 0; layer < 2; ++layer) {
        const float* W  = layer ? W2  : W1;
        const float* as = layer ? as2 : as1;
        const float* ad = layer ? ad2 : ad1;
        const float* bb = layer ? b2  : b1;

        gemm64_wmma<<<gemmBlocks, b256, 0, stream>>>(hin, W, nullptr, hx, N, 0);
        rowprep<<<(N + 255) / 256, b256, 0, stream>>>(hx, as, ad, ssrc, sdst, mx, ssum, agg, N);
        edge_alpha<<<(E + 255) / 256, b256, 0, stream>>>(src, dst, ssrc, sdst, ebuf, mx, E);
        edge_exp<<<(E + 255) / 256, b256, 0, stream>>>(dst, ebuf, mx, ssum, E);
        edge_scatter<<<(E + 7) / 8, b256, 0, stream>>>(src, dst, ebuf, ssum, hx, agg, E);
        finalize<<<((size_t)N * DD + 255) / 256, b256, 0, stream>>>(hin, agg, bb, out, N * DD);
        hin = out;   // layer 2 consumes layer 1's result in-place from d_out
    }
}